// Radon_43190191129082
// MI455X (gfx1250) — compile-verified
//
#include <hip/hip_runtime.h>
#include <stdint.h>

// Radon transform, reference-equivalent:
//   S = 725, pad_before = 106, center (S-1)/2 = 362
//   sample of output (i,j) at image coords:
//     fx = c*(j-362) + s*(i-362) + 256
//     fy = -s*(j-362) + c*(i-362) + 256
//   out[n, j, t] = sum_i bilinear(img, fx, fy)  (zero outside [0,512)^2)

#define S_DIM    725
#define W_IMG    512
#define N_ANG    180
#define JT       32          // output columns per workgroup tile
#define IT       32          // output rows per chunk
#define NJT      ((S_DIM + JT - 1) / JT)   // 23
#define MAXBB    48          // max bbox edge: 31*sqrt(2) + 4 margin -> <= 48

typedef unsigned int u32x4 __attribute__((ext_vector_type(4)));
typedef int          i32x8 __attribute__((ext_vector_type(8)));
typedef int          i32x4 __attribute__((ext_vector_type(4)));

// Accurate sin/cos for th in [0, pi): 2-term Cody-Waite reduction around pi/2
// + degree-7/8 minimax polys (~1 ulp on [-pi/4, pi/4]).  Avoids libm's
// Payne-Hanek huge-argument path (hundreds of dead SALU instructions).
__device__ __forceinline__ void sincos_small(float th, float* sp, float* cp) {
  const float TWO_OVER_PI = 0.636619772f;
  const float PIO2_HI     = 1.57079637e+00f;   // 0x3FC90FDB
  const float PIO2_LO     = -4.37113883e-08f;  // pi/2 - PIO2_HI
  float qf = floorf(fmaf(th, TWO_OVER_PI, 0.5f));   // q in {0,1,2}
  int   q  = (int)qf;
  float r  = fmaf(-qf, PIO2_HI, th);
  r        = fmaf(-qf, PIO2_LO, r);
  float z  = r * r;
  // sin(r) ~= r + r*z*ps
  float ps = fmaf(z, fmaf(z, -1.9515296e-4f, 8.3321608e-3f), -1.6666654e-1f);
  float sinr = fmaf(r * z, ps, r);
  // cos(r) ~= 1 - z/2 + z*z*pc
  float pc = fmaf(z, fmaf(z, 2.4433157e-5f, -1.3887316e-3f), 4.1666646e-2f);
  float cosr = fmaf(z * z, pc, fmaf(z, -0.5f, 1.0f));
  float s0 = (q & 1) ? cosr : sinr;
  float c0 = (q & 1) ? sinr : cosr;
  *sp = (q & 2)        ? -s0 : s0;
  *cp = ((q + 1) & 2)  ? -c0 : c0;
}

__global__ __launch_bounds__(256) void radon_tdm_kernel(const float* __restrict__ x,
                                                        float* __restrict__ out) {
  __shared__ float tile[MAXBB * MAXBB];   // 9 KB source window
  __shared__ float red[8][JT];            // cross-wave reduction

  const int bid = blockIdx.x;
  const int jt  = bid % NJT;
  const int t   = (bid / NJT) % N_ANG;
  const int n   = bid / (NJT * N_ANG);

  const int tx = threadIdx.x;             // column within tile
  const int ty = threadIdx.y;             // wave id (blockDim = 32x8)

  const float th = (float)t * 0.017453292519943295f;
  float c, s;
  sincos_small(th, &s, &c);

  const int   j0 = jt * JT;
  const int   j  = j0 + tx;
  const float u  = (float)(j - 362);

  const float*   img     = x + (size_t)n * (W_IMG * W_IMG);
  // generic LDS pointer: low 32 bits are the LDS byte offset (aperture in high bits)
  const unsigned lds_off = (unsigned)(uintptr_t)&tile[0];

  float acc = 0.0f;

  for (int i0 = 0; i0 < S_DIM; i0 += IT) {
    // ---- bounding box of source samples for this (j-tile, i-chunk) ----
    const float u0 = (float)(j0 - 362),      u1 = (float)(j0 + JT - 1 - 362);
    const float v0 = (float)(i0 - 362),      v1 = (float)(i0 + IT - 1 - 362);
    float fx00 =  c * u0 + s * v0, fx01 =  c * u0 + s * v1;
    float fx10 =  c * u1 + s * v0, fx11 =  c * u1 + s * v1;
    float fy00 = -s * u0 + c * v0, fy01 = -s * u0 + c * v1;
    float fy10 = -s * u1 + c * v0, fy11 = -s * u1 + c * v1;
    float fxmin = fminf(fminf(fx00, fx01), fminf(fx10, fx11)) + 256.0f;
    float fxmax = fmaxf(fmaxf(fx00, fx01), fmaxf(fx10, fx11)) + 256.0f;
    float fymin = fminf(fminf(fy00, fy01), fminf(fy10, fy11)) + 256.0f;
    float fymax = fmaxf(fmaxf(fy00, fy01), fmaxf(fy10, fy11)) + 256.0f;
    int xlo = (int)floorf(fxmin) - 1;     // 1-px safety margin each side
    int xhi = (int)floorf(fxmax) + 2;     // inclusive; covers floor(fx)+1
    int ylo = (int)floorf(fymin) - 1;
    int yhi = (int)floorf(fymax) + 2;
    int xs = max(xlo, 0), xe = min(xhi + 1, W_IMG);   // clamp to image
    int ys = max(ylo, 0), ye = min(yhi + 1, W_IMG);
    int w  = xe - xs,     h  = ye - ys;               // <= 48 x 48

    __syncthreads();                      // previous chunk's LDS reads done
    if (ty == 0 && w > 0 && h > 0) {
      // ---- Tensor Data Mover: 2-D tile (w x h) of the 512x512 tensor -> LDS ----
      uint64_t ga = (uint64_t)(uintptr_t)img + (uint64_t)(ys * W_IMG + xs) * 4ull;
      u32x4 g0;
      g0[0] = 1u;                                   // count=1 valid descriptor
      g0[1] = lds_off;                              // lds_addr (bytes)
      g0[2] = (unsigned)ga;                         // global_addr[31:0]
      g0[3] = (unsigned)((ga >> 32) & 0x1FFFFFFull) // global_addr[56:32]
            | (2u << 30);                           // type = 2 ("image")
      i32x8 g1;
      g1[0] = (2 << 16);                            // data_size = 4 bytes
      g1[1] = (int)((unsigned)W_IMG << 16);         // tensor_dim0[15:0]
      g1[2] = (int)(((unsigned)W_IMG >> 16) | ((unsigned)W_IMG << 16)); // dim0 hi | dim1 lo
      g1[3] = (int)(((unsigned)W_IMG >> 16) | ((unsigned)w << 16));     // dim1 hi | tile_dim0
      g1[4] = h;                                    // tile_dim1 (tile_dim2 = 0)
      g1[5] = W_IMG;                                // tensor_dim0_stride[31:0]
      g1[6] = 0;
      g1[7] = 0;
      i32x4 g2 = {0, 0, 0, 0};
      i32x4 g3 = {0, 0, 0, 0};
      i32x8 g4 = {0, 0, 0, 0, 0, 0, 0, 0};          // extra descriptor words (6-arg form)
      __builtin_amdgcn_tensor_load_to_lds(g0, g1, g2, g3, g4, 0);
      __builtin_amdgcn_s_wait_tensorcnt(0);
    }
    __syncthreads();                      // LDS window ready for all waves

    if (j < S_DIM) {
      for (int rr = ty; rr < IT; rr += 8) {
        int i = i0 + rr;
        if (i >= S_DIM) break;
        float v   = (float)(i - 362);
        float fx  =  c * u + s * v + 256.0f;
        float fy  = -s * u + c * v + 256.0f;
        float fx0 = floorf(fx), fy0 = floorf(fy);
        int   ix0 = (int)fx0,   iy0 = (int)fy0;
        float wx1 = fx - fx0,   wy1 = fy - fy0;
        float wx0 = 1.0f - wx1, wy0 = 1.0f - wy1;
        int   lx  = ix0 - xs,   ly  = iy0 - ys;      // window-local coords
        bool vx0 = (ix0     >= 0) && (ix0     < W_IMG);
        bool vx1 = (ix0 + 1 >= 0) && (ix0 + 1 < W_IMG);
        bool vy0 = (iy0     >= 0) && (iy0     < W_IMG);
        bool vy1 = (iy0 + 1 >= 0) && (iy0 + 1 < W_IMG);
        float p00 = (vx0 && vy0) ? tile[ly * w + lx]           : 0.0f;
        float p01 = (vx1 && vy0) ? tile[ly * w + lx + 1]       : 0.0f;
        float p10 = (vx0 && vy1) ? tile[(ly + 1) * w + lx]     : 0.0f;
        float p11 = (vx1 && vy1) ? tile[(ly + 1) * w + lx + 1] : 0.0f;
        acc += (p00 * wx0 + p01 * wx1) * wy0 + (p10 * wx0 + p11 * wx1) * wy1;
      }
    }
  }

  red[ty][tx] = acc;
  __syncthreads();
  if (ty == 0 && j < S_DIM) {
    float r = 0.0f;
#pragma unroll
    for (int k = 0; k < 8; ++k) r += red[k][tx];
    out[((size_t)n * S_DIM + j) * N_ANG + t] = r;
  }
}

extern "C" void kernel_launch(void* const* d_in, const int* in_sizes, int n_in,
                              void* d_out, int out_size, void* d_ws, size_t ws_size,
                              hipStream_t stream) {
  const float* x   = (const float*)d_in[0];
  float*       out = (float*)d_out;
  const int nb = in_sizes[0] / (W_IMG * W_IMG);      // batch*channels = 2
  dim3 grid(NJT * N_ANG * nb);
  dim3 block(JT, 8);                                  // 256 threads = 8 wave32
  hipLaunchKernelGGL(radon_tdm_kernel, grid, block, 0, stream, x, out);
}